// OmDetV2Turbo_56573309223026
// MI455X (gfx1250) — compile-verified
//
#include <hip/hip_runtime.h>
#include <hip/hip_bf16.h>
#include <math.h>

// ---------------------------------------------------------------------------
// OmDetV2Turbo post-process for MI455X (gfx1250, wave32)
//   K1: per-batch top-900 of 1.8M logits (monotonic-key histogram select +
//       LDS bitonic sort), decode boxes, sigmoid only on winners.
//   K2: 900x900 class-aware NMS; pairwise union's (area_i + area_j) term is
//       computed on the matrix pipe via V_WMMA_F32_16X16X4_F32 outer-sum.
// ---------------------------------------------------------------------------

typedef __attribute__((ext_vector_type(2))) float v2f;
typedef __attribute__((ext_vector_type(8))) float v8f;

#define BATCH   16
#define NPROP   900
#define NCLS    2000
#define NLOGITS (NPROP * NCLS)     // 1,800,000 per batch
#define CAP     4096               // candidate buffer (pow2 for bitonic)
#define TPB     1024               // 32 waves (wave32)
#define NT      57                 // ceil(912/16) tiles along each axis
#define SUPW    29                 // u32 words per suppress-matrix row
#define IMG_WH  640.0f

// Monotonic float -> uint key (descending key order == descending float order)
__device__ __forceinline__ unsigned orderKey(float x) {
  unsigned u = __float_as_uint(x);
  return (u & 0x80000000u) ? ~u : (u | 0x80000000u);
}
__device__ __forceinline__ float invKey(unsigned key) {
  unsigned u = (key & 0x80000000u) ? (key ^ 0x80000000u) : ~key;
  return __uint_as_float(u);
}

// ---------------------------------------------------------------------------
// Kernel 1: top-900 selection per batch + box decode
// ---------------------------------------------------------------------------
__global__ __launch_bounds__(TPB)
void omdet_topk_kernel(const float* __restrict__ box_cls,
                       const float* __restrict__ box_pred,
                       float* __restrict__ wsBox,     // [B,900,4] xyxy*640
                       float* __restrict__ wsScore,   // [B,900]
                       int*   __restrict__ wsLab) {   // [B,900]
  __shared__ unsigned hist[1024];
  __shared__ unsigned ckey[CAP];
  __shared__ unsigned cidx[CAP];
  __shared__ unsigned s_cnt;
  __shared__ unsigned s_tbin;

  const int b   = blockIdx.x;
  const int tid = threadIdx.x;
  const float* __restrict__ logits = box_cls + (size_t)b * NLOGITS;

  for (int i = tid; i < 1024; i += TPB) hist[i] = 0u;
  if (tid == 0) s_cnt = 0u;
  __syncthreads();

  // Pass 1: histogram of top 10 key bits (streams 115MB/16 per block; the
  // global_prefetch_b8 keeps the HBM pipe full ahead of the stride).
  for (int i = tid; i < NLOGITS; i += TPB) {
    __builtin_prefetch(logits + i + 8 * TPB, 0, 1);
    unsigned key = orderKey(logits[i]);
    atomicAdd(&hist[key >> 22], 1u);
  }
  __syncthreads();

  // Find bin containing the 900th largest key (serial scan of 1024 bins).
  if (tid == 0) {
    unsigned cum = 0; int bsel = 0;
    for (int bin = 1023; bin >= 0; --bin) {
      cum += hist[bin];
      if (cum >= (unsigned)NPROP) { bsel = bin; break; }
    }
    s_tbin = (unsigned)bsel;
  }
  __syncthreads();
  const unsigned thr = s_tbin << 22;

  // Pass 2: collect all candidates >= bin lower edge (~2.5K expected).
  for (int i = tid; i < NLOGITS; i += TPB) {
    unsigned key = orderKey(logits[i]);
    if (key >= thr) {
      unsigned pos = atomicAdd(&s_cnt, 1u);
      if (pos < CAP) { ckey[pos] = key; cidx[pos] = (unsigned)i; }
    }
  }
  __syncthreads();
  unsigned n = s_cnt; if (n > CAP) n = CAP;
  for (unsigned i = n + tid; i < CAP; i += TPB) { ckey[i] = 0u; cidx[i] = 0u; }
  __syncthreads();

  // Bitonic sort descending by key over CAP elements in LDS.
  for (int k = 2; k <= CAP; k <<= 1) {
    for (int j = k >> 1; j > 0; j >>= 1) {
      for (int i = tid; i < CAP; i += TPB) {
        int ixj = i ^ j;
        if (ixj > i) {
          bool desc = ((i & k) == 0);
          unsigned a = ckey[i], c = ckey[ixj];
          if (desc ? (a < c) : (a > c)) {
            unsigned ia = cidx[i], ic = cidx[ixj];
            ckey[i] = c;  ckey[ixj] = a;
            cidx[i] = ic; cidx[ixj] = ia;
          }
        }
      }
      __syncthreads();
    }
  }

  // Decode the 900 winners only (sigmoid here, not on all 28.8M logits).
  if (tid < NPROP) {
    unsigned key = ckey[tid], idx = cidx[tid];
    float logit = invKey(key);
    float score = 1.0f / (1.0f + expf(-logit));
    int label = (int)(idx % (unsigned)NCLS);
    int bi    = (int)(idx / (unsigned)NCLS);
    const float* bp = box_pred + ((size_t)b * NPROP + bi) * 4;
    float cx = bp[0], cy = bp[1], w = bp[2], h = bp[3];
    size_t o = ((size_t)b * NPROP + tid) * 4;
    wsBox[o + 0] = (cx - 0.5f * w) * IMG_WH;
    wsBox[o + 1] = (cy - 0.5f * h) * IMG_WH;
    wsBox[o + 2] = (cx + 0.5f * w) * IMG_WH;
    wsBox[o + 3] = (cy + 0.5f * h) * IMG_WH;
    wsScore[(size_t)b * NPROP + tid] = score;
    wsLab  [(size_t)b * NPROP + tid] = label;
  }
}

// ---------------------------------------------------------------------------
// Kernel 2: pairwise-IoU suppress matrix (WMMA outer-sum for union term),
// sequential greedy keep scan, clip + pack outputs.
// d_out layout (floats): out5 [B,900,5] | labels [B,900] | keep [B,900]
// ---------------------------------------------------------------------------
__global__ __launch_bounds__(TPB)
void omdet_nms_kernel(const float* __restrict__ wsBox,
                      const float* __restrict__ wsScore,
                      const int*   __restrict__ wsLab,
                      float* __restrict__ out) {
  extern __shared__ char smem[];
  float*    sbox  = (float*)smem;                 // 900*4
  float*    sarea = sbox + NPROP * 4;             // 900
  int*      slab  = (int*)(sarea + NPROP);        // 900
  int*      skeep = slab + NPROP;                 // 900
  unsigned* sup   = (unsigned*)(skeep + NPROP);   // 900*SUPW bitmask

  const int b   = blockIdx.x;
  const int tid = threadIdx.x;

  for (int i = tid; i < NPROP; i += TPB) {
    size_t o = ((size_t)b * NPROP + i) * 4;
    float x1 = wsBox[o + 0], y1 = wsBox[o + 1];
    float x2 = wsBox[o + 2], y2 = wsBox[o + 3];
    sbox[i * 4 + 0] = x1; sbox[i * 4 + 1] = y1;
    sbox[i * 4 + 2] = x2; sbox[i * 4 + 3] = y2;
    sarea[i] = fmaxf(x2 - x1, 0.0f) * fmaxf(y2 - y1, 0.0f);
    slab[i]  = wsLab[(size_t)b * NPROP + i];
    skeep[i] = 1;
  }
  for (int i = tid; i < NPROP * SUPW; i += TPB) sup[i] = 0u;
  __syncthreads();

  // Tile the 900x900 IoU matrix into 57x57 16x16 tiles, one tile per wave
  // iteration. Union term (area_i + area_j) via the f32 matrix pipe:
  //   A(16x4) = [area_i, 1, 0, 0], B(4x16) = [1; area_j; 0; 0]  =>  D = outer sum.
  const int wave = tid >> 5;
  const int lane = tid & 31;
  const int nw   = TPB >> 5;
  for (int tt = wave; tt < NT * NT; tt += nw) {
    int ti = tt / NT, tj = tt % NT;
    v2f av, bv;
    if (lane < 16) {
      int gi = ti * 16 + lane;                  // A: lanes 0-15 hold K=0,1
      av.x = (gi < NPROP) ? sarea[gi] : 0.0f;   //   K=0 -> area_i
      av.y = 1.0f;                              //   K=1 -> 1
      int gj = tj * 16 + lane;                  // B: lanes 0-15 hold K=0,1
      bv.x = 1.0f;                              //   K=0 row -> 1
      bv.y = (gj < NPROP) ? sarea[gj] : 0.0f;   //   K=1 row -> area_j
    } else {                                    // K=2,3 halves are zero
      av.x = 0.0f; av.y = 0.0f; bv.x = 0.0f; bv.y = 0.0f;
    }
    v8f acc = {};
    acc = __builtin_amdgcn_wmma_f32_16x16x4_f32(
        /*neg_a=*/false, av, /*neg_b=*/false, bv,
        /*c_mod=*/(short)0, acc, /*reuse_a=*/false, /*reuse_b=*/false);

    // C/D layout: VGPR r, lanes 0-15 -> M=r, N=lane; lanes 16-31 -> M=r+8.
    #pragma unroll
    for (int r = 0; r < 8; ++r) {
      int M = r + ((lane >> 4) << 3);
      int N = lane & 15;
      int gi = ti * 16 + M, gj = tj * 16 + N;
      if (gi < NPROP && gj < NPROP && slab[gi] == slab[gj]) {
        float ix1 = fmaxf(sbox[gi * 4 + 0], sbox[gj * 4 + 0]);
        float iy1 = fmaxf(sbox[gi * 4 + 1], sbox[gj * 4 + 1]);
        float ix2 = fminf(sbox[gi * 4 + 2], sbox[gj * 4 + 2]);
        float iy2 = fminf(sbox[gi * 4 + 3], sbox[gj * 4 + 3]);
        float inter = fmaxf(ix2 - ix1, 0.0f) * fmaxf(iy2 - iy1, 0.0f);
        float uni   = fmaxf(acc[r] - inter, 1e-9f);  // area_i+area_j - inter
        if (inter > 0.5f * uni) {                    // IoU > NMS_THRESH
          atomicOr(&sup[gi * SUPW + (gj >> 5)], 1u << (gj & 31));
        }
      }
    }
  }
  __syncthreads();

  // Greedy sequential suppression (mirrors reference fori_loop).
  for (int i = 0; i < NPROP; ++i) {
    if (skeep[i]) {
      for (int j = tid; j < NPROP; j += TPB) {
        if (j > i && ((sup[i * SUPW + (j >> 5)] >> (j & 31)) & 1u)) skeep[j] = 0;
      }
    }
    __syncthreads();
  }

  // Clip, mask, pack outputs.
  if (tid < NPROP) {
    float kf = skeep[tid] ? 1.0f : 0.0f;
    float x1 = fminf(fmaxf(sbox[tid * 4 + 0], 0.0f), IMG_WH);
    float y1 = fminf(fmaxf(sbox[tid * 4 + 1], 0.0f), IMG_WH);
    float x2 = fminf(fmaxf(sbox[tid * 4 + 2], 0.0f), IMG_WH);
    float y2 = fminf(fmaxf(sbox[tid * 4 + 3], 0.0f), IMG_WH);
    float s  = wsScore[(size_t)b * NPROP + tid];
    size_t o5 = ((size_t)b * NPROP + tid) * 5;
    out[o5 + 0] = x1 * kf;
    out[o5 + 1] = y1 * kf;
    out[o5 + 2] = x2 * kf;
    out[o5 + 3] = y2 * kf;
    out[o5 + 4] = s * kf;
    const size_t labBase  = (size_t)BATCH * NPROP * 5;
    const size_t keepBase = labBase + (size_t)BATCH * NPROP;
    out[labBase  + (size_t)b * NPROP + tid] = (float)slab[tid];
    out[keepBase + (size_t)b * NPROP + tid] = kf;
  }
}

// ---------------------------------------------------------------------------
extern "C" void kernel_launch(void* const* d_in, const int* in_sizes, int n_in,
                              void* d_out, int out_size, void* d_ws, size_t ws_size,
                              hipStream_t stream) {
  const float* box_cls  = (const float*)d_in[0];   // [16,900,2000] f32
  const float* box_pred = (const float*)d_in[1];   // [16,900,4]    f32
  float* out = (float*)d_out;

  // Workspace carve-out: boxes | scores | labels  (~346 KB)
  float* wsBox   = (float*)d_ws;                          // 16*900*4
  float* wsScore = wsBox + (size_t)BATCH * NPROP * 4;     // 16*900
  int*   wsLab   = (int*)(wsScore + (size_t)BATCH * NPROP);

  omdet_topk_kernel<<<BATCH, TPB, 0, stream>>>(box_cls, box_pred,
                                               wsBox, wsScore, wsLab);

  size_t smem = (size_t)(NPROP * 4 + NPROP) * sizeof(float)   // sbox + sarea
              + (size_t)(2 * NPROP) * sizeof(int)             // slab + skeep
              + (size_t)NPROP * SUPW * sizeof(unsigned);      // suppress bits
  omdet_nms_kernel<<<BATCH, TPB, smem, stream>>>(wsBox, wsScore, wsLab, out);
}